// GaussianSplatModel_86105504350734
// MI455X (gfx1250) — compile-verified
//
#include <hip/hip_runtime.h>
#include <hip/hip_bf16.h>

// ---------------------------------------------------------------------------
// GaussianSplat + FFT-blur for MI455X (gfx1250).
//
// splat   : 4M global_atomic_add_f32 into 256^3 f32 volume (L2-resident, 64MB)
// blur    : exp(-2pi^2 s^2 |f|^2) mask == separable *circular* convolution
//           with g[n] = IDFT(exp(-2pi^2 s^2 (k/N)^2)). Each axis pass is a
//           GEMM with the symmetric circulant matrix C[i][j] = g[(i-j) mod N],
//           executed with V_WMMA_F32_16X16X4_F32 (f32 precision matches the
//           reference's f32 FFT).
// ---------------------------------------------------------------------------

typedef __attribute__((ext_vector_type(2))) float v2f;
typedef __attribute__((ext_vector_type(8))) float v8f;

#define GRID_N   256
#define VOL_N    (256u * 256u * 256u)   // 16,777,216 floats
#define TWO_PI   6.28318530717958647692f

// ---------------------------------------------------------------- zero volume
__global__ __launch_bounds__(256) void zero_kernel(float4* __restrict__ p) {
    p[(size_t)blockIdx.x * 256 + threadIdx.x] = make_float4(0.f, 0.f, 0.f, 0.f);
}

// --------------------------------------------------- 1D periodic blur kernel
// g[n] = (1/N) * sum_k exp(-2 pi^2 sigma^2 (fftfreq_k)^2) * cos(2 pi k n / N)
// (exact inverse DFT of the reference's per-axis frequency mask)
__global__ __launch_bounds__(256) void build_g_kernel(const float* __restrict__ sigma_param,
                                                      float* __restrict__ g) {
    const int n = threadIdx.x;
    const float sigma = fmaxf(sigma_param[0], 0.f);
    const float cexp  = -2.0f * (float)(M_PI * M_PI) * sigma * sigma;
    float s = 0.f;
    for (int k = 0; k < GRID_N; ++k) {
        const float fk = (float)(k < 128 ? k : k - 256) * (1.0f / 256.0f);
        const float mask = __expf(cexp * fk * fk);
        const int   ph   = (k * n) & 255;              // exact phase mod 2*pi
        s += mask * __cosf((float)ph * (TWO_PI / 256.0f));
    }
    g[n] = s * (1.0f / 256.0f);
}

// ------------------------------------------------- circulant matrix (symmetric)
__global__ __launch_bounds__(256) void build_C_kernel(const float* __restrict__ g,
                                                      float* __restrict__ Cm) {
    const int i = blockIdx.x, j = threadIdx.x;
    Cm[i * GRID_N + j] = g[(i - j) & 255];
}

// ------------------------------------------------------------------ splatting
__global__ __launch_bounds__(256) void splat_kernel(const float* __restrict__ means,
                                                    const float* __restrict__ weights,
                                                    float* __restrict__ vol, int n) {
    const int p = blockIdx.x * 256 + threadIdx.x;
    if (p >= n) return;
    const float w  = fmaxf(weights[p], 0.f);           // relu
    const float gx = fmaf(means[3 * p + 0], 128.f, 128.f);
    const float gy = fmaf(means[3 * p + 1], 128.f, 128.f);
    const float gz = fmaf(means[3 * p + 2], 128.f, 128.f);
    const float fx = floorf(gx), fy = floorf(gy), fz = floorf(gz);
    const int bx = (int)fx, by = (int)fy, bz = (int)fz;
    const float rx = gx - fx, ry = gy - fy, rz = gz - fz;
    const float wx[2] = {1.f - rx, rx};
    const float wy[2] = {1.f - ry, ry};
    const float wz[2] = {1.f - rz, rz};
#pragma unroll
    for (int o = 0; o < 8; ++o) {
        const int ox = o & 1, oy = (o >> 1) & 1, oz = (o >> 2) & 1;
        const int ix = bx + ox, iy = by + oy, iz = bz + oz;
        if ((unsigned)ix < 256u && (unsigned)iy < 256u && (unsigned)iz < 256u) {
            const size_t a = ((size_t)ix << 16) | ((size_t)iy << 8) | (size_t)iz;
            atomicAdd(&vol[a], wx[ox] * wy[oy] * wz[oz] * w);
        }
    }
}

// ---------------------------------------------------------------- WMMA GEMM
// One axis pass: OUT[i, n] = sum_j Cm[i,j] * IN[j, n]
//   element (j, n) of IN at  base(block) + j*SJ + n_local*SC
//   SJ = stride along the convolved axis, SC = stride between panel columns.
// Block = 256 threads = 8 waves. Each block owns 16 columns x all 256 rows;
// wave w accumulates rows [32w, 32w+32) as two 16x16 WMMA tiles sharing one
// B fragment. The 256x16 input panel is staged once in LDS (pad 17 -> the
// lane stride is coprime with 64 banks: conflict-free).
#define PAD 17

template <int SJ, int SC>
__global__ __launch_bounds__(256) void conv_gemm(const float* __restrict__ in,
                                                 const float* __restrict__ Cm,
                                                 float* __restrict__ out) {
    __shared__ float lds[GRID_N * PAD];
    const int tid = threadIdx.x;
    const int n0  = blockIdx.x * 16;

    size_t base;
    if constexpr (SJ == 256) {           // axis 1: columns are (z, x) pairs
        base = (size_t)(n0 >> 8) * 65536 + (size_t)(n0 & 255);
    } else {                             // axis 0 (SC=1) or axis 2 (SC=256)
        base = (size_t)n0 * SC;
    }

    // cooperative panel load: fast lane index follows whichever stride is 1
#pragma unroll
    for (int t = 0; t < 16; ++t) {
        const int idx = tid + 256 * t;
        int j, c;
        if constexpr (SJ == 1) { j = idx & 255; c = idx >> 8; }
        else                   { c = idx & 15;  j = idx >> 4; }
        lds[j * PAD + c] = in[base + (size_t)j * SJ + (size_t)c * SC];
    }
    __syncthreads();

    const int lane = tid & 31;
    const int wave = tid >> 5;
    const int lr   = lane & 15;          // A: M row / B,D: N col
    const int koff = (lane >> 4) << 1;   // K sub-block per half-wave: {0,2}
    const int i0   = wave * 32;

    v8f acc0 = {};
    v8f acc1 = {};
    for (int k = 0; k < GRID_N; k += 4) {
        const int kk = k + koff;
        v2f a0, a1, b;
        // Cm is symmetric -> read A fragments row-major (coalesced):
        // A0(m, kk) = Cm[i0+m, kk] = Cm[kk, i0+m]
        a0.x = Cm[(kk    ) * GRID_N + i0 + lr];
        a0.y = Cm[(kk + 1) * GRID_N + i0 + lr];
        a1.x = Cm[(kk    ) * GRID_N + i0 + 16 + lr];
        a1.y = Cm[(kk + 1) * GRID_N + i0 + 16 + lr];
        // B(kk, n) from the LDS panel
        b.x = lds[(kk    ) * PAD + lr];
        b.y = lds[(kk + 1) * PAD + lr];
        acc0 = __builtin_amdgcn_wmma_f32_16x16x4_f32(false, a0, false, b,
                                                     (short)0, acc0, false, false);
        acc1 = __builtin_amdgcn_wmma_f32_16x16x4_f32(false, a1, false, b,
                                                     (short)0, acc1, false, false);
    }

    // D layout: VGPR v -> M = v (+8 upper half-wave), N = lane & 15
    const int    rowadd = (lane >> 4) << 3;
    const size_t cbase  = base + (size_t)lr * SC;
#pragma unroll
    for (int v = 0; v < 8; ++v) {
        out[cbase + (size_t)(i0 + rowadd + v) * SJ]      = acc0[v];
        out[cbase + (size_t)(i0 + 16 + rowadd + v) * SJ] = acc1[v];
    }
}

// ---------------------------------------------------------------------------
extern "C" void kernel_launch(void* const* d_in, const int* in_sizes, int n_in,
                              void* d_out, int out_size, void* d_ws, size_t ws_size,
                              hipStream_t stream) {
    const float* means       = (const float*)d_in[0];
    const float* weights     = (const float*)d_in[1];
    const float* sigma_param = (const float*)d_in[2];
    float* out = (float*)d_out;

    // workspace: [volume 64MB][g 1KB][Cm 256KB]
    float* vol = (float*)d_ws;
    float* g   = vol + VOL_N;
    float* Cm  = g + GRID_N;

    const int npts = in_sizes[1];

    zero_kernel   <<<VOL_N / (4 * 256), 256, 0, stream>>>((float4*)vol);
    build_g_kernel<<<1, 256, 0, stream>>>(sigma_param, g);
    build_C_kernel<<<GRID_N, 256, 0, stream>>>(g, Cm);
    splat_kernel  <<<(npts + 255) / 256, 256, 0, stream>>>(means, weights, vol, npts);

    // three axis passes, ping-pong vol <-> out (out holds the final result)
    conv_gemm<65536, 1><<<4096, 256, 0, stream>>>(vol, Cm, out);  // axis 0 (z)
    conv_gemm<  256, 1><<<4096, 256, 0, stream>>>(out, Cm, vol);  // axis 1 (y)
    conv_gemm<    1, 256><<<4096, 256, 0, stream>>>(vol, Cm, out);// axis 2 (x)
}